// Pairwise_50199577756233
// MI455X (gfx1250) — compile-verified
//
#include <hip/hip_runtime.h>

typedef __attribute__((ext_vector_type(2))) float v2f;
typedef __attribute__((ext_vector_type(8))) float v8f;

// LDS (address_space(3)) typed pointers: keeps all shared-memory addressing
// 32-bit and avoids generic->LDS addrspacecast null checks in codegen.
typedef __attribute__((address_space(3))) float lds_f;
typedef __attribute__((address_space(3))) v2f   lds_v2f;

#define TS   34                 // tile row stride (floats), conflict-free & 8B-aligned
#define TSZ  (16 * TS)          // 544 floats per 16x32 tile
#define WSZ  (32 * TS)          // 1088 floats per 32x32 weight matrix (padded)
#define ISC  0.1767766952966369f   // 1/sqrt(32)
#define IS3  0.5773502691896258f   // 1/sqrt(3)

__device__ __forceinline__ v8f zero8() {
  v8f z;
#pragma unroll
  for (int i = 0; i < 8; ++i) z[i] = 0.f;
  return z;
}

// [16,32] @ [32,32] via 16x v_wmma_f32_16x16x4_f32.
// X: LDS tile [16][TS].  W: LDS matrix, W[n*TS+k] = B[k][n].
// Loads all fragments first, then issues WMMAs back-to-back (2 acc chains).
__device__ __forceinline__ void mm(const lds_f* X, const lds_f* W,
                                   v8f& c0, v8f& c1, int lm, int kb) {
  const lds_f* xr = X + lm * TS + kb;
  const lds_f* w0 = W + lm * TS + kb;
  const lds_f* w1 = W + (lm + 16) * TS + kb;
  v2f a[8], b0[8], b1[8];
#pragma unroll
  for (int k = 0; k < 8; ++k) {
    a[k]  = *(const lds_v2f*)(xr + k * 4);
    b0[k] = *(const lds_v2f*)(w0 + k * 4);
    b1[k] = *(const lds_v2f*)(w1 + k * 4);
  }
#pragma unroll
  for (int k = 0; k < 8; ++k) {
    c0 = __builtin_amdgcn_wmma_f32_16x16x4_f32(false, a[k], false, b0[k], (short)0, c0, false, false);
    c1 = __builtin_amdgcn_wmma_f32_16x16x4_f32(false, a[k], false, b1[k], (short)0, c1, false, false);
  }
}

// Store C/D accumulator layout (VGPR j: M=j+8*half, N=lane&15) to a [16][TS] tile.
__device__ __forceinline__ void store_tile(lds_f* T, const v8f& a, const v8f& b,
                                           int lm, int mb) {
#pragma unroll
  for (int j = 0; j < 8; ++j) {
    T[(mb + j) * TS + lm]      = a[j];
    T[(mb + j) * TS + lm + 16] = b[j];
  }
}

// Gated resblock on one 16-row tile group. Weights pre-transposed & pre-scaled by ISC.
// Out tiles may alias in tiles (in-place safe). Tb/Tv are scratch tiles.
__device__ __forceinline__ void resblock(
    const lds_f* Si, const lds_f* V0i, const lds_f* V1i, const lds_f* V2i,
    lds_f* So, lds_f* V0o, lds_f* V1o, lds_f* V2o,
    const lds_f* W, lds_f* Tb, lds_f* Tv,
    int lm, int kb, int mb) {
  v8f s1a = zero8(), s1b = zero8();
  mm(Si, W + 0 * WSZ, s1a, s1b, lm, kb);
  store_tile(Tb, s1a, s1b, lm, mb);                 // Tb = s1
  v8f ga = zero8(), gb = zero8();
  mm(Tb, W + 2 * WSZ, ga, gb, lm, kb);              // gate pre-activation
#pragma unroll
  for (int i = 0; i < 8; ++i) {
    ga[i]  = 1.f / (1.f + __expf(-ga[i]));
    gb[i]  = 1.f / (1.f + __expf(-gb[i]));
    s1a[i] = s1a[i] / (1.f + __expf(-s1a[i]));      // silu(s1)
    s1b[i] = s1b[i] / (1.f + __expf(-s1b[i]));
  }
  store_tile(Tb, s1a, s1b, lm, mb);                 // Tb = s2
  v8f sa = zero8(), sb = zero8();
  mm(Tb, W + 3 * WSZ, sa, sb, lm, kb);
  mm(Si, W + 5 * WSZ, sa, sb, lm, kb);
  store_tile(So, sa, sb, lm, mb);
  const lds_f* Vi[3] = {V0i, V1i, V2i};
  lds_f* Vo[3] = {V0o, V1o, V2o};
#pragma unroll
  for (int x = 0; x < 3; ++x) {
    v8f va = zero8(), vb = zero8();
    mm(Vi[x], W + 1 * WSZ, va, vb, lm, kb);
#pragma unroll
    for (int i = 0; i < 8; ++i) { va[i] *= ga[i]; vb[i] *= gb[i]; }
    store_tile(Tv, va, vb, lm, mb);                 // v2[x]
    v8f oa = zero8(), ob = zero8();
    mm(Tv, W + 4 * WSZ, oa, ob, lm, kb);
    mm(Vi[x], W + 6 * WSZ, oa, ob, lm, kb);
    store_tile(Vo[x], oa, ob, lm, mb);
  }
}

// Copy nmat 32x32 matrices to LDS, padded to TS; optional transpose; scale folded in.
__device__ void load_weights(lds_f* dst, const float* src, int nmat, bool transpose, float scale) {
  for (int idx = threadIdx.x; idx < nmat * 1024; idx += blockDim.x) {
    int m = idx >> 10, r = idx & 1023, row = r >> 5, col = r & 31;
    float v = transpose ? src[m * 1024 + col * 32 + row] : src[m * 1024 + row * 32 + col];
    dst[m * WSZ + row * TS + col] = v * scale;
  }
}

// ---------------- Kernel 1: per-node precompute for off-diagonal branch ----------
// table per node (512 floats): [0]cs [32]cv(x,y,z) [128]rs [160]rv [256]sr0 [288]vr1 [384]vr2 [480]sr3
#define NPW 8
__global__ void __launch_bounds__(256) node_pre_kernel(
    const float* __restrict__ ns, const float* __restrict__ nv,
    const float* __restrict__ tpw, const float* __restrict__ resw,
    float* __restrict__ table, int nGroups) {
  extern __shared__ float smem_raw[];
  lds_f* smem = (lds_f*)smem_raw;
  lds_f* W = smem;                      // [0..6]=rw0^T, [7..13]=rw2^T, [14..17]=tp_w[1]
  lds_f* tiles = smem + 18 * WSZ;
  load_weights(W,            resw + 0 * 7 * 1024, 7, true,  ISC);
  load_weights(W + 7 * WSZ,  resw + 2 * 7 * 1024, 7, true,  ISC);
  load_weights(W + 14 * WSZ, tpw  + 1 * 4 * 1024, 4, false, ISC);
  __syncthreads();
  int wave = threadIdx.x >> 5, lane = threadIdx.x & 31;
  int g = blockIdx.x * NPW + wave;
  if (g >= nGroups) return;
  lds_f* T  = tiles + wave * 10 * TSZ;
  lds_f *S = T, *V0 = T + TSZ, *V1 = T + 2 * TSZ, *V2 = T + 3 * TSZ;
  lds_f *C0 = T + 4 * TSZ, *C1 = T + 5 * TSZ, *C2 = T + 6 * TSZ, *C3 = T + 7 * TSZ;
  lds_f *Tb = T + 8 * TSZ, *Tv = T + 9 * TSZ;
  int lm = lane & 15, kb = (lane >> 4) * 2, mb = (lane >> 4) * 8;
  int n0 = g * 16, e = lane >> 1, c0 = (lane & 1) * 16;
  const float* ps = ns + (size_t)(n0 + e) * 32 + c0;
  const float* pv = nv + (size_t)(n0 + e) * 96 + (size_t)c0 * 3;
#pragma unroll
  for (int i = 0; i < 16; ++i) {
    S [e * TS + c0 + i] = ps[i];
    V0[e * TS + c0 + i] = pv[i * 3 + 0];
    V1[e * TS + c0 + i] = pv[i * 3 + 1];
    V2[e * TS + c0 + i] = pv[i * 3 + 2];
  }
  // cs/cv = resblock(node, rw0)
  resblock(S, V0, V1, V2, C0, C1, C2, C3, W, Tb, Tv, lm, kb, mb);
#pragma unroll
  for (int i = 0; i < 16; ++i) {
    int c = c0 + i; float* t = table + (size_t)(n0 + e) * 512;
    t[c] = C0[e*TS+c]; t[32+c] = C1[e*TS+c]; t[64+c] = C2[e*TS+c]; t[96+c] = C3[e*TS+c];
  }
  // rs/rv = resblock(node, rw2)
  resblock(S, V0, V1, V2, C0, C1, C2, C3, W + 7 * WSZ, Tb, Tv, lm, kb, mb);
#pragma unroll
  for (int i = 0; i < 16; ++i) {
    int c = c0 + i; float* t = table + (size_t)(n0 + e) * 512;
    t[128+c] = C0[e*TS+c]; t[160+c] = C1[e*TS+c]; t[192+c] = C2[e*TS+c]; t[224+c] = C3[e*TS+c];
  }
  // tp_w[1] right-side precompute (sr0, vr1, vr2, sr3) straight from accumulators
  {
    const lds_f* TW = W + 14 * WSZ;
    const lds_f* Vv[3] = {V0, V1, V2};
    auto flush = [&](const v8f& a, const v8f& b, int off) {
#pragma unroll
      for (int j = 0; j < 8; ++j) {
        float* t = table + (size_t)(n0 + mb + j) * 512 + off;
        t[lm] = a[j]; t[lm + 16] = b[j];
      }
    };
    v8f a = zero8(), b = zero8();
    mm(S, TW + 0 * WSZ, a, b, lm, kb); flush(a, b, 256);
#pragma unroll
    for (int x = 0; x < 3; ++x) { a = zero8(); b = zero8(); mm(Vv[x], TW + 1 * WSZ, a, b, lm, kb); flush(a, b, 288 + 32 * x); }
#pragma unroll
    for (int x = 0; x < 3; ++x) { a = zero8(); b = zero8(); mm(Vv[x], TW + 2 * WSZ, a, b, lm, kb); flush(a, b, 384 + 32 * x); }
    a = zero8(); b = zero8();
    mm(S, TW + 3 * WSZ, a, b, lm, kb); flush(a, b, 480);
  }
}

// ---------------- Kernel 2: fully fused diagonal branch --------------------------
#define NDW 4
__global__ void __launch_bounds__(128) diag_kernel(
    const float* __restrict__ ns, const float* __restrict__ nv,
    const float* __restrict__ tpw, const float* __restrict__ resw,
    float* __restrict__ dg_s, float* __restrict__ dg_v, int nGroups) {
  extern __shared__ float smem_raw[];
  lds_f* smem = (lds_f*)smem_raw;
  lds_f* W = smem;                      // [0..6]rw3^T [7..13]rw4^T [14..20]rw5^T [21..24]tp_w[0]
  lds_f* tiles = smem + 25 * WSZ;
  load_weights(W,            resw + 3 * 7 * 1024, 7, true,  ISC);
  load_weights(W + 7 * WSZ,  resw + 4 * 7 * 1024, 7, true,  ISC);
  load_weights(W + 14 * WSZ, resw + 5 * 7 * 1024, 7, true,  ISC);
  load_weights(W + 21 * WSZ, tpw,                 4, false, ISC);
  __syncthreads();
  int wave = threadIdx.x >> 5, lane = threadIdx.x & 31;
  int g = blockIdx.x * NDW + wave;
  if (g >= nGroups) return;
  lds_f* T = tiles + wave * 16 * TSZ;
  lds_f *S = T, *V0 = T + TSZ, *V1 = T + 2 * TSZ, *V2 = T + 3 * TSZ;
  lds_f* P[8];
#pragma unroll
  for (int i = 0; i < 8; ++i) P[i] = T + (4 + i) * TSZ;
  lds_f *D0 = T + 12 * TSZ, *D1 = T + 13 * TSZ, *D2 = T + 14 * TSZ, *D3 = T + 15 * TSZ;
  int lm = lane & 15, kb = (lane >> 4) * 2, mb = (lane >> 4) * 8;
  int n0 = g * 16, e = lane >> 1, c0 = (lane & 1) * 16;
  const float* ps = ns + (size_t)(n0 + e) * 32 + c0;
  const float* pv = nv + (size_t)(n0 + e) * 96 + (size_t)c0 * 3;
#pragma unroll
  for (int i = 0; i < 16; ++i) {
    S [e * TS + c0 + i] = ps[i];
    V0[e * TS + c0 + i] = pv[i * 3 + 0];
    V1[e * TS + c0 + i] = pv[i * 3 + 1];
    V2[e * TS + c0 + i] = pv[i * 3 + 2];
  }
  // ds/dv = resblock(node, rw3)
  resblock(S, V0, V1, V2, D0, D1, D2, D3, W, P[0], P[1], lm, kb, mb);
  // tp_w[0] right-side precompute into P0..P7
  {
    const lds_f* TW = W + 21 * WSZ;
    const lds_f* Vv[3] = {V0, V1, V2};
    v8f a = zero8(), b = zero8();
    mm(S, TW + 0 * WSZ, a, b, lm, kb); store_tile(P[0], a, b, lm, mb);
#pragma unroll
    for (int x = 0; x < 3; ++x) { a = zero8(); b = zero8(); mm(Vv[x], TW + 1 * WSZ, a, b, lm, kb); store_tile(P[1 + x], a, b, lm, mb); }
#pragma unroll
    for (int x = 0; x < 3; ++x) { a = zero8(); b = zero8(); mm(Vv[x], TW + 2 * WSZ, a, b, lm, kb); store_tile(P[4 + x], a, b, lm, mb); }
    a = zero8(); b = zero8();
    mm(S, TW + 3 * WSZ, a, b, lm, kb); store_tile(P[7], a, b, lm, mb);
  }
  // tp combine (in place over D)
#pragma unroll
  for (int i = 0; i < 16; ++i) {
    int idx = e * TS + c0 + i;
    float ds = D0[idx], dx = D1[idx], dy = D2[idx], dz = D3[idx];
    float ts = ds * P[0][idx] + (dx * P[1][idx] + dy * P[2][idx] + dz * P[3][idx]) * IS3;
    D1[idx] = ds * P[4][idx] + dx * P[7][idx];
    D2[idx] = ds * P[5][idx] + dy * P[7][idx];
    D3[idx] = ds * P[6][idx] + dz * P[7][idx];
    D0[idx] = ts;
  }
  // res_pair (rw4) in place, then res_res (rw5) to P4..7
  resblock(D0, D1, D2, D3, D0, D1, D2, D3, W + 7 * WSZ, P[0], P[1], lm, kb, mb);
  resblock(S, V0, V1, V2, P[4], P[5], P[6], P[7], W + 14 * WSZ, P[2], P[3], lm, kb, mb);
#pragma unroll
  for (int i = 0; i < 16; ++i) {
    int c = c0 + i, idx = e * TS + c;
    size_t node = (size_t)(n0 + e);
    dg_s[node * 32 + c] = D0[idx] + P[4][idx];
    float* ov = dg_v + (node * 32 + c) * 3;
    ov[0] = D1[idx] + P[5][idx];
    ov[1] = D2[idx] + P[6][idx];
    ov[2] = D3[idx] + P[7][idx];
  }
}

// ---------------- Kernel 3: per-edge branch (gathers hit L2-resident table) ------
#define NEW 8
__global__ void __launch_bounds__(256) edge_kernel(
    const long long* __restrict__ ei, const float* __restrict__ table,
    const float* __restrict__ resw,
    float* __restrict__ off_s, float* __restrict__ off_v,
    int E, int nGroups) {
  extern __shared__ float smem_raw[];
  lds_f* smem = (lds_f*)smem_raw;
  lds_f* W = smem;                      // rw1^T (7 matrices)
  lds_f* tiles = smem + 7 * WSZ;
  load_weights(W, resw + 1 * 7 * 1024, 7, true, ISC);
  __syncthreads();
  int wave = threadIdx.x >> 5, lane = threadIdx.x & 31;
  int g = blockIdx.x * NEW + wave;
  if (g >= nGroups) return;
  lds_f* T = tiles + wave * 6 * TSZ;
  lds_f *Xs = T, *X0 = T + TSZ, *X1 = T + 2 * TSZ, *X2 = T + 3 * TSZ;
  lds_f *Tb = T + 4 * TSZ, *Tv = T + 5 * TSZ;
  int lm = lane & 15, kb = (lane >> 4) * 2, mb = (lane >> 4) * 8;
  int e0 = g * 16, e = lane >> 1, c0 = (lane & 1) * 16;
  long long sn = ei[e0 + e];
  long long dn = ei[(size_t)E + e0 + e];
  const float* pa = table + (size_t)sn * 512;   // src precompute (sr0/vr1/vr2/sr3)
  const float* pb = table + (size_t)dn * 512;   // dst precompute (cs/cv/rs/rv)
#pragma unroll
  for (int i = 0; i < 16; ++i) {
    int c = c0 + i, idx = e * TS + c;
    float cs = pb[c], cvx = pb[32 + c], cvy = pb[64 + c], cvz = pb[96 + c];
    float sr0 = pa[256 + c];
    float w1x = pa[288 + c], w1y = pa[320 + c], w1z = pa[352 + c];
    float w2x = pa[384 + c], w2y = pa[416 + c], w2z = pa[448 + c];
    float sr3 = pa[480 + c];
    Xs[idx] = cs * sr0 + (cvx * w1x + cvy * w1y + cvz * w1z) * IS3;
    X0[idx] = cs * w2x + cvx * sr3;
    X1[idx] = cs * w2y + cvy * sr3;
    X2[idx] = cs * w2z + cvz * sr3;
  }
  resblock(Xs, X0, X1, X2, Xs, X0, X1, X2, W, Tb, Tv, lm, kb, mb);
#pragma unroll
  for (int i = 0; i < 16; ++i) {
    int c = c0 + i, idx = e * TS + c;
    size_t eg = (size_t)(e0 + e);
    off_s[eg * 32 + c] = Xs[idx] + pb[128 + c];
    float* ov = off_v + (eg * 32 + c) * 3;
    ov[0] = X0[idx] + pb[160 + c];
    ov[1] = X1[idx] + pb[192 + c];
    ov[2] = X2[idx] + pb[224 + c];
  }
}

extern "C" void kernel_launch(void* const* d_in, const int* in_sizes, int n_in,
                              void* d_out, int out_size, void* d_ws, size_t ws_size,
                              hipStream_t stream) {
  const float*     node_s = (const float*)d_in[0];
  const float*     node_v = (const float*)d_in[1];
  const float*     tp_w   = (const float*)d_in[2];
  const float*     res_w  = (const float*)d_in[3];
  const long long* eidx   = (const long long*)d_in[4];
  int N = in_sizes[0] / 32;
  int E = in_sizes[4] / 2;

  float* out   = (float*)d_out;
  float* dg_s  = out;
  float* dg_v  = out + (size_t)N * 32;
  float* off_s = out + (size_t)N * 128;
  float* off_v = off_s + (size_t)E * 32;
  float* table = (float*)d_ws;          // N * 512 floats (~102 MB, L2-resident)

  int gN = N / 16;                      // N = 50000 -> 3125 groups
  int gE = E / 16;                      // E = 800000 -> 50000 groups

  size_t shm1 = (size_t)(18 * WSZ + NPW * 10 * TSZ) * 4;
  size_t shm2 = (size_t)(25 * WSZ + NDW * 16 * TSZ) * 4;
  size_t shm3 = (size_t)(7 * WSZ + NEW * 6 * TSZ) * 4;
  (void)hipFuncSetAttribute((const void*)node_pre_kernel,
                            hipFuncAttributeMaxDynamicSharedMemorySize, (int)shm1);
  (void)hipFuncSetAttribute((const void*)diag_kernel,
                            hipFuncAttributeMaxDynamicSharedMemorySize, (int)shm2);
  (void)hipFuncSetAttribute((const void*)edge_kernel,
                            hipFuncAttributeMaxDynamicSharedMemorySize, (int)shm3);

  node_pre_kernel<<<(gN + NPW - 1) / NPW, 256, shm1, stream>>>(
      node_s, node_v, tp_w, res_w, table, gN);
  diag_kernel<<<(gN + NDW - 1) / NDW, 128, shm2, stream>>>(
      node_s, node_v, tp_w, res_w, dg_s, dg_v, gN);
  edge_kernel<<<(gE + NEW - 1) / NEW, 256, shm3, stream>>>(
      eidx, table, res_w, off_s, off_v, E, gE);
}